// Model_45870250721572
// MI455X (gfx1250) — compile-verified
//
#include <hip/hip_runtime.h>
#include <hip/hip_bf16.h>

// EA-LSTM for MI455X (gfx1250), fp32 WMMA (V_WMMA_F32_16X16X4_F32)
// + async global->LDS staging of the shared A tile.
//
// Sizes: B=256, T=365, Dd=32, Ds=27, H=256, 3H=768.
// d_out layout: out[256] ++ h_n[B*T*H] ++ c_n[B*T*H]
// d_ws: i_gate[B*H] floats (256 KB).

typedef __attribute__((ext_vector_type(2))) float v2f;
typedef __attribute__((ext_vector_type(8))) float v8f;
typedef __attribute__((ext_vector_type(4))) int   v4i;

// Exact pointer types for __builtin_amdgcn_global_load_async_to_lds_b128:
// arg0: v4i addrspace(1)* (global), arg1: v4i addrspace(3)* (LDS).
typedef v4i __attribute__((address_space(1)))* gv4i_p;
typedef v4i __attribute__((address_space(3)))* lv4i_p;

#define EB   256
#define ET   365
#define EDD  32
#define EH   256
#define EG3  768
#define EK   (EDD + EH)      // 288 total K
#define LDSP 292             // padded row pitch (floats): 36*lid mod 64 hits 16 distinct banks

#if defined(__has_builtin)
#if __has_builtin(__builtin_amdgcn_global_load_async_to_lds_b128)
#define HAVE_ASYNC_LDS 1
#endif
#endif

__device__ __forceinline__ gv4i_p as_global_b128(const float* p) {
    return (gv4i_p)(unsigned long long)(uintptr_t)p;   // AS1 shares generic 64-bit repr
}
__device__ __forceinline__ lv4i_p as_lds_b128(const float* p) {
    return (lv4i_p)(unsigned)(uintptr_t)p;             // low 32 bits = LDS byte offset
}

__device__ __forceinline__ float sigmoid_f(float x) {
    return 1.0f / (1.0f + __expf(-x));
}

// 3 WMMAs (f/o/g gate tiles) sharing one A fragment; B rows kr,kr+1 of wr0/wr1.
__device__ __forceinline__ void fog_wmma(
    const v2f a, const float* wr0, const float* wr1, const int ncol,
    v8f& accf, v8f& acco, v8f& accg)
{
    v2f bf; bf.x = wr0[ncol];          bf.y = wr1[ncol];
    v2f bo; bo.x = wr0[EH + ncol];     bo.y = wr1[EH + ncol];
    v2f bg; bg.x = wr0[2 * EH + ncol]; bg.y = wr1[2 * EH + ncol];
    accf = __builtin_amdgcn_wmma_f32_16x16x4_f32(
        false, a, false, bf, (short)0, accf, false, false);
    acco = __builtin_amdgcn_wmma_f32_16x16x4_f32(
        false, a, false, bo, (short)0, acco, false, false);
    accg = __builtin_amdgcn_wmma_f32_16x16x4_f32(
        false, a, false, bg, (short)0, accg, false, false);
}

// ---------------------------------------------------------------------------
// Kernel 1: i_gate = sigmoid(x_s @ W_sh + b_s)   [B,H], K=27 (tiny, VALU)
// ---------------------------------------------------------------------------
__global__ __launch_bounds__(EH) void ealstm_igate(
    const float* __restrict__ x_s,    // [B,27]
    const float* __restrict__ w_sh,   // [27,H]
    const float* __restrict__ b_s,    // [H]
    float* __restrict__ i_gate)       // [B,H]
{
    const int b = blockIdx.x;
    const int h = threadIdx.x;
    const float* xs = x_s + b * 27;
    float acc = b_s[h];
#pragma unroll
    for (int k = 0; k < 27; ++k)
        acc = fmaf(xs[k], w_sh[k * EH + h], acc);
    i_gate[b * EH + h] = sigmoid_f(acc);
}

// ---------------------------------------------------------------------------
// Kernel 2: one recurrence step.
//   gates[256,768] = [x_t | h_{t-1}] (256x288) @ [W_ih; W_hh] (288x768) + bias
//   c_t = sigmoid(f)*c_{t-1} + i_gate*tanh(g);  h_t = sigmoid(o)*tanh(c_t)
//
// 16 blocks x 512 threads (16 waves). Block = one 16-batch-row tile (tm),
// wave = one 16-col tile within H (tn). The block's shared A tile
// (16 x 288 fp32) is staged to LDS once per step via async B128 copies
// (ASYNCcnt + barrier); each wave then computes f/o/g 16x16 tiles with
// three V_WMMA_F32_16X16X4_F32 per K-chunk, sharing one ds_load_b64 A frag.
// The K loop is split into the x segment (W_ih) and the h segment (W_hh)
// so the dependent 64-chunk loop is branch-free with constant-stride bumps.
// ---------------------------------------------------------------------------
__global__ __launch_bounds__(512) void ealstm_step(
    const float* __restrict__ x_d,    // [B,T,Dd]
    const float* __restrict__ w_ih,   // [Dd,3H]
    const float* __restrict__ w_hh,   // [H,3H]
    const float* __restrict__ bias,   // [3H]
    const float* __restrict__ i_gate, // [B,H]
    float* __restrict__ h_n,          // [B,T,H]
    float* __restrict__ c_n,          // [B,T,H]
    int t)
{
    __shared__ float smem[16 * LDSP];          // A tile: cols [0,32)=x, [32,288)=h

    const int tid  = threadIdx.x;
    const int lane = tid & 31;
    const int wave = tid >> 5;                 // 0..15 = tn
    const int tm   = blockIdx.x;               // 0..15 batch row-tile
    const int tn   = wave;

    // ---- Stage A tile into LDS (coalesced, no redundancy) ----
#if HAVE_ASYNC_LDS
    {
        // x part: 16 rows x 32 floats = 128 B128 copies (threads 0..127)
        if (tid < 128) {
            const int row = tid >> 3;
            const int col = (tid & 7) * 4;
            const float* g = x_d + ((size_t)(tm * 16 + row) * ET + t) * EDD + col;
            __builtin_amdgcn_global_load_async_to_lds_b128(
                as_global_b128(g), as_lds_b128(&smem[row * LDSP + col]), 0, 0);
        }
        // h part: 16 rows x 256 floats = 1024 B128 copies (2 per thread)
        if (t > 0) {
            const int row = tid >> 5;
            const int cb  = (tid & 31) * 8;
#pragma unroll
            for (int j = 0; j < 2; ++j) {
                const int col = cb + j * 4;
                const float* g = h_n + ((size_t)(tm * 16 + row) * ET + (t - 1)) * EH + col;
                __builtin_amdgcn_global_load_async_to_lds_b128(
                    as_global_b128(g), as_lds_b128(&smem[row * LDSP + EDD + col]), 0, 0);
            }
        }
#if __has_builtin(__builtin_amdgcn_s_wait_asynccnt)
        __builtin_amdgcn_s_wait_asynccnt(0);
#else
        asm volatile("s_wait_asynccnt 0x0" ::: "memory");
#endif
    }
#else
    {
        // Fallback: synchronous staged copy.
        for (int idx = tid; idx < 16 * EDD; idx += 512) {
            const int row = idx >> 5, col = idx & 31;
            smem[row * LDSP + col] =
                x_d[((size_t)(tm * 16 + row) * ET + t) * EDD + col];
        }
        if (t > 0) {
            for (int idx = tid; idx < 16 * EH; idx += 512) {
                const int row = idx >> 8, col = idx & 255;
                smem[row * LDSP + EDD + col] =
                    h_n[((size_t)(tm * 16 + row) * ET + (t - 1)) * EH + col];
            }
        }
    }
#endif
    __syncthreads();

    const int half = lane >> 4;               // 0: K={0,1}, 1: K={2,3}
    const int lid  = lane & 15;
    const int koff = half * 2;
    const int ncol = tn * 16 + lid;           // output column within [0,H)

    v8f accf = {}; v8f acco = {}; v8f accg = {};

    const float* arow = &smem[lid * LDSP];    // this lane's A row in LDS

    // Segment 1: x_t @ W_ih  (8 chunks, K = 0..31)
#pragma unroll 2
    for (int k = 0; k < EDD; k += 4) {
        const v2f a = *(const v2f*)&arow[k + koff];
        const float* wr0 = w_ih + (size_t)(k + koff) * EG3;
        fog_wmma(a, wr0, wr0 + EG3, ncol, accf, acco, accg);
    }
    // Segment 2: h_{t-1} @ W_hh  (64 chunks, K = 32..287), skipped at t==0.
    if (t > 0) {
#pragma unroll 2
        for (int k = 0; k < EH; k += 4) {
            const v2f a = *(const v2f*)&arow[EDD + k + koff];
            const float* wr0 = w_hh + (size_t)(k + koff) * EG3;
            fog_wmma(a, wr0, wr0 + EG3, ncol, accf, acco, accg);
        }
    }

    // Epilogue: C/D layout — VGPR i, lanes 0-15: M=i, N=lane; lanes 16-31: M=8+i.
    const int   hcol = tn * 16 + lid;
    const float bf_  = bias[hcol];
    const float bo_  = bias[EH + hcol];
    const float bg_  = bias[2 * EH + hcol];
#pragma unroll
    for (int i = 0; i < 8; ++i) {
        const int m = tm * 16 + i + half * 8;   // batch index
        const size_t cur = ((size_t)m * ET + t) * EH + hcol;
        const float f = accf[i] + bf_;
        const float o = acco[i] + bo_;
        const float g = accg[i] + bg_;
        const float cprev = (t == 0) ? 0.0f
                          : c_n[((size_t)m * ET + (t - 1)) * EH + hcol];
        const float ig = i_gate[m * EH + hcol];
        const float c1 = sigmoid_f(f) * cprev + ig * tanhf(g);
        const float h1 = sigmoid_f(o) * tanhf(c1);
        h_n[cur] = h1;
        c_n[cur] = c1;
    }
}

// ---------------------------------------------------------------------------
// Kernel 3: out[b] = h_n[b, T-1, :] @ fc_w + fc_b
// ---------------------------------------------------------------------------
__global__ __launch_bounds__(EB) void ealstm_fc(
    const float* __restrict__ h_n,    // [B,T,H]
    const float* __restrict__ fc_w,   // [H]
    const float* __restrict__ fc_b,   // [1]
    float* __restrict__ out)          // [B]
{
    const int b = threadIdx.x;
    const float* hp = h_n + ((size_t)b * ET + (ET - 1)) * EH;
    float s = 0.0f;
#pragma unroll 8
    for (int i = 0; i < EH; ++i)
        s = fmaf(hp[i], fc_w[i], s);
    out[b] = s + fc_b[0];
}

// ---------------------------------------------------------------------------
extern "C" void kernel_launch(void* const* d_in, const int* in_sizes, int n_in,
                              void* d_out, int out_size, void* d_ws, size_t ws_size,
                              hipStream_t stream) {
    (void)in_sizes; (void)n_in; (void)out_size; (void)ws_size;
    const float* x_d   = (const float*)d_in[0];   // [B,T,Dd]
    const float* x_s   = (const float*)d_in[1];   // [B,Ds]
    const float* w_ih  = (const float*)d_in[2];   // [Dd,3H]
    const float* w_hh  = (const float*)d_in[3];   // [H,3H]
    const float* w_sh  = (const float*)d_in[4];   // [Ds,H]
    const float* bias  = (const float*)d_in[5];   // [3H]
    const float* b_s   = (const float*)d_in[6];   // [H]
    const float* fc_w  = (const float*)d_in[7];   // [H,1]
    const float* fc_b  = (const float*)d_in[8];   // [1]

    float* out = (float*)d_out;                        // [B]
    float* h_n = out + EB;                             // [B,T,H]
    float* c_n = h_n + (size_t)EB * ET * EH;           // [B,T,H]
    float* i_gate = (float*)d_ws;                      // [B,H]

    // Static input gate (once).
    ealstm_igate<<<EB, EH, 0, stream>>>(x_s, w_sh, b_s, i_gate);

    // 365 dependent recurrence steps; h/c history lives in d_out so step t
    // reads step t-1's slice. All nodes recorded into the captured graph.
    for (int t = 0; t < ET; ++t) {
        ealstm_step<<<16, 512, 0, stream>>>(x_d, w_ih, w_hh, bias, i_gate,
                                            h_n, c_n, t);
    }

    // Final FC on the last hidden state.
    ealstm_fc<<<1, EB, 0, stream>>>(h_n, fc_w, fc_b, out);
}